// CrossModalAttention_54606214201757
// MI455X (gfx1250) — compile-verified
//
#include <hip/hip_runtime.h>
#include <math.h>

// ---------------------------------------------------------------------------
// CDNA5 (gfx1250) fused multi-head cross-attention, exact-f32 WMMA path.
//   1) q/k/v projections: GEMM + bias via V_WMMA_F32_16X16X4_F32
//   2) flash-attention (online softmax, scores never hit HBM)
//   3) output projection: GEMM + bias
// Data movement: Tensor Data Mover (tensor_load_to_lds) with double-buffered
// LDS tiles + hardware row padding; s_wait_tensorcnt for completion.
// Fallback to cooperative VGPR staging when the TDM builtin is unavailable.
// ---------------------------------------------------------------------------

typedef __attribute__((ext_vector_type(2))) float v2f;
typedef __attribute__((ext_vector_type(8))) float v8f;
typedef __attribute__((ext_vector_type(4))) unsigned int u32x4;
typedef __attribute__((ext_vector_type(4))) int i32x4;
typedef __attribute__((ext_vector_type(8))) int i32x8;

#define DIM      1024
#define HEADS    16
#define HEAD_DIM 64
#define TSEQ     2048
#define BATCH    2
#define MROWS    (BATCH * TSEQ)   // 4096

#if __has_builtin(__builtin_amdgcn_tensor_load_to_lds) && \
    __has_builtin(__builtin_amdgcn_s_wait_tensorcnt)
#define USE_TDM 1
#else
#define USE_TDM 0
#endif

static __device__ __forceinline__ v8f wmma_f32(v2f a, v2f b, v8f c) {
    // 8 args: (neg_a, A, neg_b, B, c_mod, C, reuse_a, reuse_b)
    return __builtin_amdgcn_wmma_f32_16x16x4_f32(
        false, a, false, b, (short)0, c, false, false);
}

#if USE_TDM
// ---------------------------------------------------------------------------
// Issue one TDM 2-D tile load: tile_h rows x tile_w dwords, row stride
// row_stride dwords, into LDS at lds_off with hardware row padding
// (pad_interval code c -> pad every 2^(c+1) dwords; pad_amount code a ->
// insert a+1 dwords). Descriptor layout per CDNA5 ISA ch.8.
// ---------------------------------------------------------------------------
static __device__ __forceinline__ void tdm_load_tile(
        unsigned lds_off, const float* gsrc,
        unsigned tile_w, unsigned tile_h, unsigned row_stride,
        unsigned pad_interval_code, unsigned pad_amount_code) {
    const unsigned long long ga = (unsigned long long)gsrc;
    u32x4 g0;
    g0.x = 1u;                                   // count=1, user mode, no gather
    g0.y = lds_off;                              // lds_addr [63:32]
    g0.z = (unsigned)ga;                         // global_addr [95:64]
    g0.w = (unsigned)((ga >> 32) & 0x01FFFFFFu)  // global_addr [120:96]
         | 0x80000000u;                          // type=2 ("image") [127:126]
    i32x8 g1;
    g1[0] = (2 << 16)                            // data_size = 4 bytes
          | (1 << 20)                            // pad_enable
          | ((int)(pad_interval_code & 7u) << 22)
          | ((int)(pad_amount_code & 0x7Fu) << 25);
    g1[1] = (int)(row_stride << 16);             // tensor_dim0[15:0]
    g1[2] = (int)((row_stride >> 16) & 0xFFFFu); // tensor_dim0 hi | tensor_dim1 lo(=0)
    g1[3] = (int)(0x10u | (tile_w << 16));       // tensor_dim1 hi (1M rows) | tile_dim0
    g1[4] = (int)tile_h;                         // tile_dim1 | tile_dim2(=0)
    g1[5] = (int)row_stride;                     // tensor_dim0_stride[31:0]
    g1[6] = 0;                                   // stride hi | tensor_dim1_stride lo
    g1[7] = 0;
    i32x4 g2 = {1, 1, 0, 0};                     // tensor_dim2/3 = 1, dims 3/4 unused
    i32x4 g3 = {0, 0, 0, 0};
#if defined(__clang_major__) && __clang_major__ >= 23
    i32x8 g4 = {};
    __builtin_amdgcn_tensor_load_to_lds(g0, g1, g2, g3, g4, 0);
#else
    __builtin_amdgcn_tensor_load_to_lds(g0, g1, g2, g3, 0);
#endif
}

// low 32 bits of a generic pointer to __shared__ == LDS byte offset
static __device__ __forceinline__ unsigned lds_offset(const void* p) {
    return (unsigned)(unsigned long long)p;
}
#endif  // USE_TDM

// ---------------------------------------------------------------------------
// GEMM: Y[M][N] = X[M][K] @ W^T + bias   (W stored [N][K] row-major)
// Block = 256 threads = 8 waves. Block tile 128(M) x 64(N). K chunks of 32,
// double-buffered in LDS via TDM. Each wave computes a 16x64 strip.
// ---------------------------------------------------------------------------
__global__ __launch_bounds__(256)
void gemm_bias_kernel(const float* __restrict__ X, const float* __restrict__ W,
                      const float* __restrict__ bias, float* __restrict__ Y,
                      int M, int N, int K) {
    __shared__ float Xs[2][128][36];   // 32 data + 4 pad dwords per row
    __shared__ float Ws[2][64][36];

    const int t    = threadIdx.x;
    const int lane = t & 31;
    const int wave = t >> 5;
    const int hh   = lane >> 4;
    const int ln   = lane & 15;
    const int m0   = blockIdx.y * 128;
    const int n0   = blockIdx.x * 64;
    const int nck  = K / 32;

    v8f acc[4] = {};

#if USE_TDM
    if (wave == 0) {   // prologue: stage chunk 0 into buffer 0
        tdm_load_tile(lds_offset(&Xs[0][0][0]), &X[(size_t)m0 * K], 32, 128, K, 4, 3);
        tdm_load_tile(lds_offset(&Ws[0][0][0]), &W[(size_t)n0 * K], 32, 64,  K, 4, 3);
    }
#endif

    for (int ck = 0; ck < nck; ++ck) {
        const int buf = ck & 1;
#if USE_TDM
        if (wave == 0) {
            if (ck + 1 < nck) {   // prefetch next chunk into other buffer
                const int kk = (ck + 1) * 32;
                tdm_load_tile(lds_offset(&Xs[buf ^ 1][0][0]),
                              &X[(size_t)m0 * K + kk], 32, 128, K, 4, 3);
                tdm_load_tile(lds_offset(&Ws[buf ^ 1][0][0]),
                              &W[(size_t)n0 * K + kk], 32, 64,  K, 4, 3);
                __builtin_amdgcn_s_wait_tensorcnt(2);  // current buffer complete
            } else {
                __builtin_amdgcn_s_wait_tensorcnt(0);
            }
        }
#else
        {
            const int kk = ck * 32;
            #pragma unroll
            for (int i = 0; i < 4; ++i) {
                int idx = t + i * 256;
                int r = idx >> 3, p = idx & 7;
                *(float4*)&Xs[buf][r][p * 4] =
                    *(const float4*)&X[(size_t)(m0 + r) * K + kk + p * 4];
            }
            #pragma unroll
            for (int i = 0; i < 2; ++i) {
                int idx = t + i * 256;
                int r = idx >> 3, p = idx & 7;
                *(float4*)&Ws[buf][r][p * 4] =
                    *(const float4*)&W[(size_t)(n0 + r) * K + kk + p * 4];
            }
        }
#endif
        __syncthreads();

        // compute: batch fragment loads (4 K-steps) then a 16-wmma burst
        #pragma unroll
        for (int kg = 0; kg < 2; ++kg) {
            v2f a[4];
            v2f bf[4][4];
            #pragma unroll
            for (int ks = 0; ks < 4; ++ks) {
                const int kb = kg * 16 + ks * 4;
                a[ks].x = Xs[buf][wave * 16 + ln][kb + 2 * hh];
                a[ks].y = Xs[buf][wave * 16 + ln][kb + 2 * hh + 1];
                #pragma unroll
                for (int nt = 0; nt < 4; ++nt) {
                    bf[ks][nt].x = Ws[buf][nt * 16 + ln][kb + hh];
                    bf[ks][nt].y = Ws[buf][nt * 16 + ln][kb + 2 + hh];
                }
            }
            #pragma unroll
            for (int ks = 0; ks < 4; ++ks)
                #pragma unroll
                for (int nt = 0; nt < 4; ++nt)
                    acc[nt] = wmma_f32(a[ks], bf[ks][nt], acc[nt]);
        }
        __syncthreads();
    }

    // epilogue: C/D layout -> row = j + 8*h, col = ln
    #pragma unroll
    for (int nt = 0; nt < 4; ++nt) {
        const int col = n0 + nt * 16 + ln;
        const float bv = bias[col];
        #pragma unroll
        for (int j = 0; j < 8; ++j) {
            const int row = m0 + wave * 16 + j + 8 * hh;
            Y[(size_t)row * N + col] = acc[nt][j] + bv;
        }
    }
}

// ---------------------------------------------------------------------------
// Flash attention over one (batch, head): block covers 128 query rows
// (8 waves x 16). K/V 16x64 tiles double-buffered in LDS via TDM; online
// softmax keeps running max/sum per row; scores never leave the WGP.
// q/k/v in [B, T, DIM] layout; head h occupies columns [h*64, h*64+64).
// ---------------------------------------------------------------------------
__global__ __launch_bounds__(256)
void flash_attn_kernel(const float* __restrict__ Q, const float* __restrict__ Kg,
                       const float* __restrict__ V, const int* __restrict__ mask,
                       float* __restrict__ O) {
    __shared__ float Ks[2][16][68];        // 64 data + 4 pad dwords per row
    __shared__ float Vs[2][16][68];
    __shared__ float Ss[8][16][17];        // per-wave score tile
    __shared__ float alpha_s[8][16];
    __shared__ float lsum_s[8][16];

    const int t    = threadIdx.x;
    const int lane = t & 31;
    const int wave = t >> 5;
    const int hh   = lane >> 4;
    const int ln   = lane & 15;
    const int b    = blockIdx.z;
    const int head = blockIdx.y;
    const int qbase = blockIdx.x * 128 + wave * 16;
    const int NKB  = TSEQ / 16;

    const float* Qb = Q  + (size_t)b * TSEQ * DIM + (size_t)head * HEAD_DIM;
    const float* Kb = Kg + (size_t)b * TSEQ * DIM + (size_t)head * HEAD_DIM;
    const float* Vb = V  + (size_t)b * TSEQ * DIM + (size_t)head * HEAD_DIM;
    const int*   Mb = mask + (size_t)b * TSEQ * TSEQ;

    // Q fragments for all 16 K-steps of head_dim=64; fold in 1/sqrt(64)
    v2f qa[16];
    #pragma unroll
    for (int s = 0; s < 16; ++s) {
        const float* qr = Qb + (size_t)(qbase + ln) * DIM + s * 4 + 2 * hh;
        qa[s].x = qr[0] * 0.125f;
        qa[s].y = qr[1] * 0.125f;
    }

    v8f o[4] = {};
    float m_run = -1e30f;   // valid in lanes 0..15 (lane == row)
    float l_run = 0.0f;

#if USE_TDM
    if (wave == 0) {   // prologue: stage K/V tile 0 into buffer 0
        tdm_load_tile(lds_offset(&Ks[0][0][0]), Kb, 64, 16, DIM, 5, 3);
        tdm_load_tile(lds_offset(&Vs[0][0][0]), Vb, 64, 16, DIM, 5, 3);
    }
#endif

    for (int kb = 0; kb < NKB; ++kb) {
        const int buf = kb & 1;
#if USE_TDM
        if (wave == 0) {
            if (kb + 1 < NKB) {
                const size_t roff = (size_t)(kb + 1) * 16 * DIM;
                tdm_load_tile(lds_offset(&Ks[buf ^ 1][0][0]), Kb + roff, 64, 16, DIM, 5, 3);
                tdm_load_tile(lds_offset(&Vs[buf ^ 1][0][0]), Vb + roff, 64, 16, DIM, 5, 3);
                __builtin_amdgcn_s_wait_tensorcnt(2);
            } else {
                __builtin_amdgcn_s_wait_tensorcnt(0);
            }
        }
#else
        {
            const int r = t >> 4, p = t & 15;
            *(float4*)&Ks[buf][r][p * 4] =
                *(const float4*)&Kb[(size_t)(kb * 16 + r) * DIM + p * 4];
            *(float4*)&Vs[buf][r][p * 4] =
                *(const float4*)&Vb[(size_t)(kb * 16 + r) * DIM + p * 4];
        }
#endif
        __syncthreads();

        // S = (Q*scale) @ K^T : 16 wmma, fragments batched 4 at a time
        v8f s_acc = {};
        #pragma unroll
        for (int sg = 0; sg < 4; ++sg) {
            v2f bf[4];
            #pragma unroll
            for (int s4 = 0; s4 < 4; ++s4) {
                const int s = sg * 4 + s4;
                bf[s4].x = Ks[buf][ln][s * 4 + hh];
                bf[s4].y = Ks[buf][ln][s * 4 + 2 + hh];
            }
            #pragma unroll
            for (int s4 = 0; s4 < 4; ++s4)
                s_acc = wmma_f32(qa[sg * 4 + s4], bf[s4], s_acc);
        }

        // clip +-100, mask -> -1e30, spill C-layout tile to LDS row-major
        #pragma unroll
        for (int j = 0; j < 8; ++j) {
            const int row = j + 8 * hh;
            float sv = fminf(fmaxf(s_acc[j], -100.0f), 100.0f);
            const int mq = Mb[(size_t)(qbase + row) * TSEQ + kb * 16 + ln];
            if (mq == 0) sv = -1e30f;
            Ss[wave][row][ln] = sv;
        }
        __syncthreads();

        // online softmax: lane r owns row r
        if (lane < 16) {
            float mx = -1e30f;
            #pragma unroll
            for (int c = 0; c < 16; ++c) mx = fmaxf(mx, Ss[wave][lane][c]);
            const float m_new = fmaxf(m_run, mx);
            const float al = __expf(m_run - m_new);
            float ps = 0.0f;
            #pragma unroll
            for (int c = 0; c < 16; ++c) {
                const float p = __expf(Ss[wave][lane][c] - m_new);
                Ss[wave][lane][c] = p;
                ps += p;
            }
            l_run = l_run * al + ps;
            m_run = m_new;
            alpha_s[wave][lane] = al;
        }
        __syncthreads();

        // O = O*alpha + P @ V : batch all fragments, then a 16-wmma burst
        #pragma unroll
        for (int nt = 0; nt < 4; ++nt)
            #pragma unroll
            for (int j = 0; j < 8; ++j)
                o[nt][j] *= alpha_s[wave][j + 8 * hh];

        {
            v2f pa[4];
            v2f vb[4][4];
            #pragma unroll
            for (int s = 0; s < 4; ++s) {
                pa[s].x = Ss[wave][ln][s * 4 + 2 * hh];
                pa[s].y = Ss[wave][ln][s * 4 + 2 * hh + 1];
                #pragma unroll
                for (int nt = 0; nt < 4; ++nt) {
                    vb[s][nt].x = Vs[buf][s * 4 + hh][nt * 16 + ln];
                    vb[s][nt].y = Vs[buf][s * 4 + 2 + hh][nt * 16 + ln];
                }
            }
            #pragma unroll
            for (int s = 0; s < 4; ++s)
                #pragma unroll
                for (int nt = 0; nt < 4; ++nt)
                    o[nt] = wmma_f32(pa[s], vb[s][nt], o[nt]);
        }
        __syncthreads();
    }

    if (lane < 16) lsum_s[wave][lane] = l_run;
    __syncthreads();

    float* Ob = O + (size_t)b * TSEQ * DIM + (size_t)head * HEAD_DIM;
    #pragma unroll
    for (int nt = 0; nt < 4; ++nt) {
        #pragma unroll
        for (int j = 0; j < 8; ++j) {
            const int row = j + 8 * hh;
            const float inv = 1.0f / lsum_s[wave][row];
            Ob[(size_t)(qbase + row) * DIM + nt * 16 + ln] = o[nt][j] * inv;
        }
    }
}

// ---------------------------------------------------------------------------
// inputs: 0 query, 1 key, 2 value, 3 mask, 4 wq, 5 wk, 6 wv, 7 wo,
//         8 bq, 9 bk, 10 bv, 11 bo
// workspace: q | k | v | attn_out, each 4096*1024 f32 (64 MB total)
// ---------------------------------------------------------------------------
extern "C" void kernel_launch(void* const* d_in, const int* in_sizes, int n_in,
                              void* d_out, int out_size, void* d_ws, size_t ws_size,
                              hipStream_t stream) {
    (void)in_sizes; (void)n_in; (void)out_size; (void)ws_size;

    const float* query = (const float*)d_in[0];
    const float* key   = (const float*)d_in[1];
    const float* value = (const float*)d_in[2];
    const int*   mask  = (const int*)  d_in[3];
    const float* wq    = (const float*)d_in[4];
    const float* wk    = (const float*)d_in[5];
    const float* wv    = (const float*)d_in[6];
    const float* wo    = (const float*)d_in[7];
    const float* bq    = (const float*)d_in[8];
    const float* bk    = (const float*)d_in[9];
    const float* bv    = (const float*)d_in[10];
    const float* bo    = (const float*)d_in[11];

    const size_t buf = (size_t)MROWS * DIM;   // 4M floats
    float* qp = (float*)d_ws;
    float* kp = qp + buf;
    float* vp = kp + buf;
    float* op = vp + buf;

    const dim3 gblk(DIM / 64, MROWS / 128);   // (16, 32)
    gemm_bias_kernel<<<gblk, 256, 0, stream>>>(query, wq, bq, qp, MROWS, DIM, DIM);
    gemm_bias_kernel<<<gblk, 256, 0, stream>>>(key,   wk, bk, kp, MROWS, DIM, DIM);
    gemm_bias_kernel<<<gblk, 256, 0, stream>>>(value, wv, bv, vp, MROWS, DIM, DIM);

    flash_attn_kernel<<<dim3(TSEQ / 128, HEADS, BATCH), 256, 0, stream>>>(
        qp, kp, vp, mask, op);

    gemm_bias_kernel<<<gblk, 256, 0, stream>>>(op, wo, bo, (float*)d_out,
                                               MROWS, DIM, DIM);
}